// Head_26723286515992
// MI455X (gfx1250) — compile-verified
//
#include <hip/hip_runtime.h>
#include <hip/hip_bf16.h>
#include <math.h>

// Problem constants (from reference)
#define BATCH   256
#define SEQ     256
#define NEMB    384
#define HEAD    64

// LDS strides (padded for bank-conflict-free, 8B-aligned access)
#define QKV_STR 66   // row stride (floats) for Q/K/V tiles in LDS
#define P_STR   18   // row stride (floats) for per-wave P-tile scratch
#define X_STR   18   // row stride (floats) of TDM-staged X panel (16 data + 2 pad)
#define NPANEL  (NEMB / 16)   // 24 K-panels of 16 columns

typedef float        v2f __attribute__((ext_vector_type(2)));
typedef float        v8f __attribute__((ext_vector_type(8)));
typedef unsigned int v4u __attribute__((ext_vector_type(4)));
typedef int          v4i __attribute__((ext_vector_type(4)));
typedef int          v8i __attribute__((ext_vector_type(8)));

__device__ __forceinline__ v8f wmma_f32(v2f a, v2f b, v8f c) {
    // V_WMMA_F32_16X16X4_F32 : D = A(16x4) * B(4x16) + C(16x16)
    return __builtin_amdgcn_wmma_f32_16x16x4_f32(
        false, a, false, b, (short)0, c, false, false);
}

// Tensor Data Mover: DMA one 256-row x 16-col f32 tile of X into LDS.
// D# per CDNA5 ISA ch.8: group0 = {flags, lds_addr, global_addr, type},
// group1 = {mask/size/pad, tensor dims, tile dims, strides}. Row padding:
// pad_interval code 3 (every 16 dwords) + pad_amount code 1 (2 dwords)
// -> LDS row stride 18 floats, bank-conflict-free & 8B aligned.
__device__ __forceinline__ void tdm_load_x_panel(const float* tile_gva,
                                                 unsigned lds_byte_off) {
    const unsigned long long va = (unsigned long long)(uintptr_t)tile_gva;
    v4u g0;
    g0.x = 1u;                                   // count=1 (valid), is_restore=0
    g0.y = lds_byte_off;                         // LDS destination (bytes)
    g0.z = (unsigned)(va & 0xFFFFFFFFu);         // global_addr[31:0]
    g0.w = (unsigned)((va >> 32) & 0x01FFFFFFu)  // global_addr[56:32]
         | (2u << 30);                           // type = 2 ("image")
    v8i g1;
    g1[0] = (2 << 16)                            // data_size = 2 (4 bytes)
          | (1 << 20)                            // pad_enable
          | (3 << 22)                            // pad_interval: 16 dwords
          | (1 << 25);                           // pad_amount: 2 dwords
    g1[1] = (int)(NEMB << 16);                   // tensor_dim0[15:0] = 384
    g1[2] = (int)(SEQ << 16);                    // dim0[31:16]=0, tensor_dim1 = 256
    g1[3] = (int)(16 << 16);                     // tensor_dim1 hi = 0, tile_dim0 = 16
    g1[4] = SEQ;                                 // tile_dim1 = 256, tile_dim2 = 0
    g1[5] = NEMB;                                // tensor_dim0_stride = 384
    g1[6] = 0;                                   // stride hi / tensor_dim1_stride lo
    g1[7] = 0;
    v4i gz4 = {0, 0, 0, 0};                      // groups 2/3 unused (2D tensor)
    v8i gz8 = {0, 0, 0, 0, 0, 0, 0, 0};
    __builtin_amdgcn_tensor_load_to_lds(g0, g1, gz4, gz4, gz8, 0);
}

__global__ void __launch_bounds__(256, 1)
attn_head_kernel(const float* __restrict__ X,
                 const float* __restrict__ Wq,
                 const float* __restrict__ Wk,
                 const float* __restrict__ Wv,
                 float* __restrict__ Out)
{
    __shared__ float Qs[SEQ * QKV_STR];
    __shared__ float Ks[SEQ * QKV_STR];
    __shared__ float Vs[SEQ * QKV_STR];
    __shared__ float Ps[8 * 16 * P_STR];      // per-wave P-tile scratch
    __shared__ float Xs[2][SEQ * X_STR];      // TDM-staged X panels (double buffer)

    const int b    = blockIdx.x;
    const int tid  = threadIdx.x;
    const int wv   = tid >> 5;      // wave id 0..7
    const int lane = tid & 31;
    const int lo   = lane & 15;     // N / M index within fragment
    const int hi   = lane >> 4;     // half-wave select
    const int dk   = hi * 2;        // K-pair offset for A/B fragments (f32 WMMA)
    const int m_of = hi * 8;        // C/D row offset for this half-wave

    const float scale = 0.05103103630798288f; // 384^-0.5 (reference scales by n_embed)
    const v8f zero8 = {0.f,0.f,0.f,0.f,0.f,0.f,0.f,0.f};

    const float* xb = X + (size_t)b * SEQ * NEMB;       // this batch's X (256x384)
    const unsigned xs_off0 = (unsigned)(uintptr_t)&Xs[0][0];
    const unsigned xs_off1 = (unsigned)(uintptr_t)&Xs[1][0];

    // ---------------- Phase 1: Q,K,V projections (TDM -> LDS -> WMMA) ----------
    // Two sweeps; in sweep si wave wv owns row strip (wv or wv+8). Each sweep
    // pipelines 24 K-panels of X through a double-buffered TDM stage.
    for (int si = 0; si < 2; ++si) {
        const int strip = si ? (wv + 8) : wv;
        const int r0 = strip * 16;

        v8f aq[4], ak[4], av[4];
        #pragma unroll
        for (int nt = 0; nt < 4; ++nt) { aq[nt] = zero8; ak[nt] = zero8; av[nt] = zero8; }

        if (wv == 0) tdm_load_x_panel(xb, xs_off0);     // prologue: panel 0

        for (int p = 0; p < NPANEL; ++p) {
            if (wv == 0) {
                if (p < NPANEL - 1) {                   // prefetch next panel
                    tdm_load_x_panel(xb + (p + 1) * 16, (p & 1) ? xs_off0 : xs_off1);
                    __builtin_amdgcn_s_wait_tensorcnt(1);   // panel p landed
                } else {
                    __builtin_amdgcn_s_wait_tensorcnt(0);
                }
            }
            __syncthreads();                            // panel p visible to all

            const float* xp = &Xs[p & 1][0];
            #pragma unroll
            for (int ks = 0; ks < 4; ++ks) {
                const v2f a = *(const v2f*)&xp[(r0 + lo) * X_STR + ks * 4 + dk];
                const int wbase = (p * 16 + ks * 4 + dk) * HEAD;
                #pragma unroll
                for (int nt = 0; nt < 4; ++nt) {
                    const int n0 = nt * 16 + lo;
                    v2f bq = { Wq[wbase + n0], Wq[wbase + HEAD + n0] };
                    v2f bk = { Wk[wbase + n0], Wk[wbase + HEAD + n0] };
                    v2f bv = { Wv[wbase + n0], Wv[wbase + HEAD + n0] };
                    aq[nt] = wmma_f32(a, bq, aq[nt]);
                    ak[nt] = wmma_f32(a, bk, ak[nt]);
                    av[nt] = wmma_f32(a, bv, av[nt]);
                }
            }
            __syncthreads();                            // done reading buf[p&1]
        }

        // Scatter C/D layout (row m = v + m_of, col n = nt*16+lo) into LDS.
        #pragma unroll
        for (int nt = 0; nt < 4; ++nt) {
            const int col = nt * 16 + lo;
            #pragma unroll
            for (int v = 0; v < 8; ++v) {
                const int row = r0 + v + m_of;
                Qs[row * QKV_STR + col] = aq[nt][v] * scale; // fold softmax scale into Q
                Ks[row * QKV_STR + col] = ak[nt][v];
                Vs[row * QKV_STR + col] = av[nt][v];
            }
        }
    }
    __syncthreads();

    // ---------------- Phase 2: causal attention ----------------
    float* Pw = Ps + wv * 16 * P_STR;

    for (int qi = 0; qi < 2; ++qi) {
        const int qtile = qi ? (15 - wv) : wv;   // {w, 15-w}: balanced causal work
        const int qr0 = qtile * 16;

        // --- S = (Q*scale) . K^T, only k-tiles <= qtile (causal) ---
        v8f S[16];
        #pragma unroll
        for (int kt = 0; kt < 16; ++kt) {
            if (kt > qtile) continue;            // wave-uniform branch, EXEC stays full
            v8f s = zero8;
            #pragma unroll
            for (int ks = 0; ks < 16; ++ks) {
                const int h = ks * 4 + dk;
                v2f a  = *(const v2f*)&Qs[(qr0 + lo) * QKV_STR + h];        // A: Q rows
                v2f bb = *(const v2f*)&Ks[(kt * 16 + lo) * QKV_STR + h];    // B: K^T cols
                s = wmma_f32(a, bb, s);
            }
            if (kt == qtile) {                   // diagonal tile: mask n > m
                #pragma unroll
                for (int v = 0; v < 8; ++v)
                    if (lo > v + m_of) s[v] = -INFINITY;
            }
            S[kt] = s;
        }

        // --- row max (rows live in 16-lane half-groups, one row per C/D vgpr) ---
        float rmax[8], rsum[8];
        #pragma unroll
        for (int v = 0; v < 8; ++v) rmax[v] = -INFINITY;
        #pragma unroll
        for (int kt = 0; kt < 16; ++kt) {
            if (kt > qtile) continue;
            #pragma unroll
            for (int v = 0; v < 8; ++v) rmax[v] = fmaxf(rmax[v], S[kt][v]);
        }
        #pragma unroll
        for (int v = 0; v < 8; ++v)
            for (int off = 8; off >= 1; off >>= 1)
                rmax[v] = fmaxf(rmax[v], __shfl_xor(rmax[v], off, 16));

        // --- exponentiate in place, accumulate row sums ---
        #pragma unroll
        for (int v = 0; v < 8; ++v) rsum[v] = 0.f;
        #pragma unroll
        for (int kt = 0; kt < 16; ++kt) {
            if (kt > qtile) continue;
            #pragma unroll
            for (int v = 0; v < 8; ++v) {
                float p = __expf(S[kt][v] - rmax[v]);
                S[kt][v] = p;
                rsum[v] += p;
            }
        }
        #pragma unroll
        for (int v = 0; v < 8; ++v)
            for (int off = 8; off >= 1; off >>= 1)
                rsum[v] += __shfl_xor(rsum[v], off, 16);

        // --- O = P . V (unnormalized), re-fragment P via per-wave LDS scratch ---
        v8f O[4];
        #pragma unroll
        for (int nt = 0; nt < 4; ++nt) O[nt] = zero8;

        #pragma unroll
        for (int kt = 0; kt < 16; ++kt) {
            if (kt > qtile) continue;
            // C/D layout -> LDS (row m = v+m_of, col n = lo)
            #pragma unroll
            for (int v = 0; v < 8; ++v)
                Pw[(v + m_of) * P_STR + lo] = S[kt][v];
            asm volatile("s_wait_dscnt 0" ::: "memory");   // order store -> A-frag loads

            #pragma unroll
            for (int ks = 0; ks < 4; ++ks) {               // K = 16 keys, 4 per WMMA
                v2f a = *(const v2f*)&Pw[lo * P_STR + ks * 4 + dk];
                #pragma unroll
                for (int nt = 0; nt < 4; ++nt) {
                    const int key = kt * 16 + ks * 4 + dk;
                    v2f bb = { Vs[key * QKV_STR + nt * 16 + lo],
                               Vs[(key + 1) * QKV_STR + nt * 16 + lo] };
                    O[nt] = wmma_f32(a, bb, O[nt]);
                }
            }
        }

        // --- normalize rows by 1/sum and write out ---
        float rinv[8];
        #pragma unroll
        for (int v = 0; v < 8; ++v) rinv[v] = 1.0f / rsum[v];

        #pragma unroll
        for (int nt = 0; nt < 4; ++nt) {
            const int col = nt * 16 + lo;
            #pragma unroll
            for (int v = 0; v < 8; ++v) {
                const int row = qr0 + v + m_of;
                Out[((size_t)b * SEQ + row) * HEAD + col] = O[nt][v] * rinv[v];
            }
        }
    }
}

extern "C" void kernel_launch(void* const* d_in, const int* in_sizes, int n_in,
                              void* d_out, int out_size, void* d_ws, size_t ws_size,
                              hipStream_t stream) {
    const float* x  = (const float*)d_in[0];
    const float* Wq = (const float*)d_in[1];
    const float* Wk = (const float*)d_in[2];
    const float* Wv = (const float*)d_in[3];
    float* out = (float*)d_out;
    attn_head_kernel<<<dim3(BATCH), dim3(256), 0, stream>>>(x, Wq, Wk, Wv, out);
}